// TopKAutoencodeInhibitor_88665304858727
// MI455X (gfx1250) — compile-verified
//
#include <hip/hip_runtime.h>
#include <math.h>

// Problem constants (from reference setup_inputs)
#define N_ROWS 8192
#define D_DIM  1024
#define M_EXP  128
#define B_DIM  64
#define K_SEL  8
#define EPS_F  1e-8f

typedef float v2f __attribute__((ext_vector_type(2)));
typedef float v8f __attribute__((ext_vector_type(8)));

// ---------------------------------------------------------------------------
// K1: energy[n,i] = sum_c h[n,i,c]^2. One wave32 per (n,i) pair; each lane
// loads a float2 (coalesced 256B per wave) and we reduce with wave shuffles.
// ---------------------------------------------------------------------------
__global__ void energy_kernel(const float* __restrict__ h,
                              float* __restrict__ energy) {
  const int wave = (int)((blockIdx.x * blockDim.x + threadIdx.x) >> 5);
  const int lane = (int)(threadIdx.x & 31);
  if (wave >= N_ROWS * M_EXP) return;
  const float2* hp = (const float2*)(h + (size_t)wave * B_DIM);
  float2 v = hp[lane];
  float e = v.x * v.x + v.y * v.y;
  #pragma unroll
  for (int off = 16; off; off >>= 1) e += __shfl_xor(e, off, 32);
  if (lane == 0) energy[wave] = e;
}

// ---------------------------------------------------------------------------
// K2: per-expert energy sums over all N rows (for balance entropy).
// One block per expert.
// ---------------------------------------------------------------------------
__global__ void expert_sum_kernel(const float* __restrict__ energy,
                                  float* __restrict__ sum_e) {
  const int i = (int)blockIdx.x;
  const int t = (int)threadIdx.x;
  float s = 0.0f;
  for (int n = t; n < N_ROWS; n += (int)blockDim.x)
    s += energy[(size_t)n * M_EXP + i];
  __shared__ float red[256];
  red[t] = s;
  __syncthreads();
  for (int off = 128; off; off >>= 1) {
    if (t < off) red[t] += red[t + off];
    __syncthreads();
  }
  if (t == 0) sum_e[i] = red[0];
}

// ---------------------------------------------------------------------------
// K3: top-8 selection per row (one wave32 per row; 128 energies = 4/lane),
// gather h_sparse to d_out, emit indices (as float, d_out dtype is f32),
// accumulate captured energy, and bin rows by expert for the GEMM pass.
// ---------------------------------------------------------------------------
__global__ void topk_kernel(const float* __restrict__ h,
                            const float* __restrict__ energy,
                            float* __restrict__ hs_out,
                            float* __restrict__ idx_out,
                            int* __restrict__ counts,
                            int* __restrict__ rows,
                            float* __restrict__ acc) {
  const int wib  = (int)(threadIdx.x >> 5);
  const int lane = (int)(threadIdx.x & 31);
  const int n = (int)blockIdx.x * ((int)blockDim.x >> 5) + wib;
  if (n >= N_ROWS) return;

  const float* er = energy + (size_t)n * M_EXP;
  float e[4];
  #pragma unroll
  for (int q = 0; q < 4; q++) e[q] = er[q * 32 + lane];

  float vals[K_SEL];
  int   idxs[K_SEL];
  #pragma unroll
  for (int j = 0; j < K_SEL; j++) {
    float v = -INFINITY;
    int   ii = 0;
    #pragma unroll
    for (int q = 0; q < 4; q++) {
      if (e[q] > v) { v = e[q]; ii = q * 32 + lane; }
    }
    #pragma unroll
    for (int off = 16; off; off >>= 1) {
      float ov = __shfl_xor(v, off, 32);
      int   oi = __shfl_xor(ii, off, 32);
      if (ov > v || (ov == v && oi < ii)) { v = ov; ii = oi; }
    }
    vals[j] = v;
    idxs[j] = ii;
    #pragma unroll
    for (int q = 0; q < 4; q++)
      if (ii == q * 32 + lane) e[q] = -INFINITY;
  }

  // Gather h_sparse (N,k,b) into d_out (float2 per lane per expert)
  #pragma unroll
  for (int j = 0; j < K_SEL; j++) {
    const float2* src =
        (const float2*)(h + ((size_t)n * M_EXP + idxs[j]) * B_DIM);
    float2 vv = src[lane];
    float2* dst = (float2*)(hs_out + ((size_t)n * K_SEL + j) * B_DIM);
    dst[lane] = vv;
  }
  if (lane < K_SEL) idx_out[(size_t)n * K_SEL + lane] = (float)idxs[lane];

  if (lane == 0) {
    float cs = 0.0f;
    #pragma unroll
    for (int j = 0; j < K_SEL; j++) cs += vals[j];
    atomicAdd(acc + 0, cs);
    #pragma unroll
    for (int j = 0; j < K_SEL; j++) {
      int pos = atomicAdd(&counts[idxs[j]], 1);
      rows[(size_t)idxs[j] * N_ROWS + pos] = n;
    }
  }
}

// ---------------------------------------------------------------------------
// K4: expert-major reconstruction GEMM with V_WMMA_F32_16X16X4_F32.
// Block = 1 wave32; grid = (D/16, m). For expert i and a 16-wide d-slab:
//   D(16x16) += A(16 gathered rows x K=4 of h) * B(K=4 x 16 d's of V)
// K loops over b=64 as 16 WMMA steps; row tiles loop over the expert's
// gathered row list. Ragged tail lanes redirect their row POINTER to a
// 64-float zero row in the workspace (single 64-bit select, no per-element
// masking, no EXEC divergence). All A loads are unconditional b64 so the
// compiler can batch 16 loads -> one wait -> 16-WMMA burst. Scatter row
// indices come from a wave shuffle instead of global re-reads. f32 WMMA
// keeps exact reference precision; the op is HBM-bound so lower precision
// buys nothing.
// ---------------------------------------------------------------------------
__global__ void recon_gemm_kernel(const float* __restrict__ h,
                                  const float* __restrict__ V,
                                  const int* __restrict__ counts,
                                  const int* __restrict__ rows,
                                  const float* __restrict__ zero_row,
                                  float* __restrict__ xhat) {
  const int lane  = (int)threadIdx.x;   // blockDim.x == 32
  const int dbase = (int)blockIdx.x * 16;
  const int i     = (int)blockIdx.y;
  const int cnt   = counts[i];
  if (cnt == 0) return;

  const int col = lane & 15;
  const int hi  = lane >> 4;            // 0: K pair {0,1}; 1: K pair {2,3}

  // Preload B fragments as float2: V[(dbase+col), i, kb*4 + 2*hi + {0,1}]
  const float2* vp2 =
      (const float2*)(V + ((size_t)(dbase + col) * M_EXP + i) * B_DIM);
  v2f bfrag[16];
  #pragma unroll
  for (int kb = 0; kb < 16; kb++) {
    float2 t = vp2[kb * 2 + hi];
    v2f b;
    b.x = t.x;
    b.y = t.y;
    bfrag[kb] = b;
  }

  const int* rowbase = rows + (size_t)i * N_ROWS;
  const int  ntiles  = (cnt + 15) >> 4;

  for (int t = 0; t < ntiles; t++) {
    const int  mrow  = lane & 15;
    const int  r     = t * 16 + mrow;
    const bool valid = (r < cnt);
    const int  rc    = valid ? r : (cnt - 1);       // clamped, in-bounds
    const int  nr    = rowbase[rc];                 // row index for M=mrow
    const float* hrow = h + ((size_t)nr * M_EXP + i) * B_DIM;
    const float2* hp2 =
        (const float2*)(valid ? hrow : zero_row);   // pointer select only

    // Unconditional, batchable A-fragment loads (16x b64 per lane).
    float2 areg[16];
    #pragma unroll
    for (int kb = 0; kb < 16; kb++) areg[kb] = hp2[kb * 2 + hi];

    v8f c = {};
    #pragma unroll
    for (int kb = 0; kb < 16; kb++) {
      v2f a;
      a.x = areg[kb].x;
      a.y = areg[kb].y;
      c = __builtin_amdgcn_wmma_f32_16x16x4_f32(
          /*neg_a=*/false, a, /*neg_b=*/false, bfrag[kb],
          /*c_mod=*/(short)0, c, /*reuse_a=*/false, /*reuse_b=*/false);
    }

    // Scatter: VGPR r8 holds D[M = r8 (+8 for upper lanes)][N = lane&15].
    // Row index for output M comes from a shuffle of nr (lane M holds it).
    #pragma unroll
    for (int r8 = 0; r8 < 8; r8++) {
      const int mo = hi ? (r8 + 8) : r8;
      const int no = __shfl(nr, mo, 32);   // executed by all lanes
      const int rr = t * 16 + mo;
      if (rr < cnt) {
        atomicAdd(&xhat[(size_t)no * D_DIM + dbase + col], c[r8]);
      }
    }
  }
}

// ---------------------------------------------------------------------------
// K5: per-row residual and reconstruction energies, accumulated globally.
// ---------------------------------------------------------------------------
__global__ void resid_kernel(const float* __restrict__ x,
                             const float* __restrict__ xhat,
                             float* __restrict__ acc) {
  const int n = (int)blockIdx.x;
  const int t = (int)threadIdx.x;
  float su = 0.0f, sr = 0.0f;
  for (int d = t; d < D_DIM; d += (int)blockDim.x) {
    const float xh = xhat[(size_t)n * D_DIM + d];
    const float r  = x[(size_t)n * D_DIM + d] - xh;
    su += r * r;
    sr += xh * xh;
  }
  #pragma unroll
  for (int off = 16; off; off >>= 1) {
    su += __shfl_xor(su, off, 32);
    sr += __shfl_xor(sr, off, 32);
  }
  __shared__ float redu[8], redr[8];
  const int w = t >> 5, lane = t & 31;
  if (lane == 0) { redu[w] = su; redr[w] = sr; }
  __syncthreads();
  if (t == 0) {
    float tu = 0.0f, tr = 0.0f;
    for (int ww = 0; ww < ((int)blockDim.x >> 5); ww++) {
      tu += redu[ww];
      tr += redr[ww];
    }
    atomicAdd(acc + 2, tu);  // uncaptured sum
    atomicAdd(acc + 1, tr);  // recon sum
  }
}

// ---------------------------------------------------------------------------
// K6: finalize scalars (single wave): balance entropy over mean expert
// energies, then the 5 scalar outputs.
// ---------------------------------------------------------------------------
__global__ void finalize_kernel(const float* __restrict__ sum_e,
                                const float* __restrict__ acc,
                                float* __restrict__ out_scalars) {
  const int lane = (int)threadIdx.x;
  float avg[4];
  float s = 0.0f;
  #pragma unroll
  for (int q = 0; q < 4; q++) {
    avg[q] = sum_e[q * 32 + lane] / (float)N_ROWS;
    s += avg[q];
  }
  #pragma unroll
  for (int off = 16; off; off >>= 1) s += __shfl_xor(s, off, 32);
  const float denom = fmaxf(s, EPS_F);
  float ent = 0.0f;
  #pragma unroll
  for (int q = 0; q < 4; q++) {
    const float p = fmaxf(avg[q] / denom, EPS_F);
    ent += p * logf(p);
  }
  #pragma unroll
  for (int off = 16; off; off >>= 1) ent += __shfl_xor(ent, off, 32);
  ent = -ent / logf((float)M_EXP);
  if (lane == 0) {
    const float cap = acc[0] / (float)N_ROWS;
    const float rec = acc[1] / (float)N_ROWS;
    const float unc = acc[2] / (float)N_ROWS;
    out_scalars[0] = cap;                       // captured_energy_proj
    out_scalars[1] = rec;                       // recon_energy
    out_scalars[2] = unc;                       // uncaptured_energy
    out_scalars[3] = ent;                       // balance_entropy
    out_scalars[4] = unc + 0.5f * (1.0f - ent); // topk_aux_loss
  }
}

// ---------------------------------------------------------------------------
// Launch. Inputs: d_in[0]=x_flat (N,D) f32, d_in[1]=h_all (N,m,b) f32,
// d_in[2]=V (D,m,b) f32, d_in[3]=k (==8). Output layout (all f32):
//   h_sparse (N*k*b) | topk_idxs as float (N*k) | 5 scalars.
// Workspace: energy (N*m) | sum_e (m) | acc (4) | counts (m) |
//            rows (m*N) | xhat (N*D) | zero_row (64)  -> ~40 MB.
// ---------------------------------------------------------------------------
extern "C" void kernel_launch(void* const* d_in, const int* in_sizes, int n_in,
                              void* d_out, int out_size, void* d_ws,
                              size_t ws_size, hipStream_t stream) {
  const float* x = (const float*)d_in[0];
  const float* h = (const float*)d_in[1];
  const float* V = (const float*)d_in[2];
  (void)in_sizes; (void)n_in; (void)out_size; (void)ws_size;

  float* energy = (float*)d_ws;                              // N*m
  float* sum_e  = energy + (size_t)N_ROWS * M_EXP;           // m
  float* acc    = sum_e + M_EXP;                             // 4
  int*   counts = (int*)(acc + 4);                           // m
  int*   rows   = counts + M_EXP;                            // m*N
  float* xhat   = (float*)(rows + (size_t)M_EXP * N_ROWS);   // N*D
  float* zrow   = xhat + (size_t)N_ROWS * D_DIM;             // B_DIM zeros

  float* out      = (float*)d_out;
  float* hs_out   = out;
  float* idx_out  = hs_out + (size_t)N_ROWS * K_SEL * B_DIM;
  float* scal_out = idx_out + (size_t)N_ROWS * K_SEL;

  // Deterministic per-call zeroing of accumulators/bins/x_hat/pad row.
  hipMemsetAsync(counts, 0, sizeof(int) * M_EXP, stream);
  hipMemsetAsync(acc, 0, sizeof(float) * 4, stream);
  hipMemsetAsync(xhat, 0, sizeof(float) * (size_t)N_ROWS * D_DIM, stream);
  hipMemsetAsync(zrow, 0, sizeof(float) * B_DIM, stream);

  // K1: energies — one wave per (n,i), 8 waves per block.
  energy_kernel<<<(N_ROWS * M_EXP) / 8, 256, 0, stream>>>(h, energy);

  // K2: per-expert sums for entropy.
  expert_sum_kernel<<<M_EXP, 256, 0, stream>>>(energy, sum_e);

  // K3: top-8 select + gather + binning — one wave per row.
  topk_kernel<<<N_ROWS / 8, 256, 0, stream>>>(h, energy, hs_out, idx_out,
                                              counts, rows, acc);

  // K4: expert-major WMMA f32 GEMM with atomic scatter into x_hat.
  recon_gemm_kernel<<<dim3(D_DIM / 16, M_EXP), 32, 0, stream>>>(
      h, V, counts, rows, zrow, xhat);

  // K5: residual / recon energy reduction.
  resid_kernel<<<N_ROWS, 256, 0, stream>>>(x, xhat, acc);

  // K6: scalars.
  finalize_kernel<<<1, 32, 0, stream>>>(sum_e, acc, scal_out);
}